// HGATConv_4346506903712
// MI455X (gfx1250) — compile-verified
//
#include <hip/hip_runtime.h>

#define N_NODES 50000
#define E_EDGES 800000
#define ET (E_EDGES + N_NODES)
#define EPSF 1e-15f
#define CLIPF (1.0f - 1e-7f)
#define MAXNF (1.0f - 4e-3f)

typedef float v2f __attribute__((ext_vector_type(2)));
typedef float v8f __attribute__((ext_vector_type(8)));

// ---- workspace layout (units: floats) ----
#define OFF_WT   0                               // 128*128 pair-interleaved W^T
#define OFF_HX   16384                           // N*128: mx, then overwritten in-place by x_t0
#define OFF_NATT (OFF_HX + N_NODES * 128)        // N*4: (ai0, ai1, aj0, aj1)
#define OFF_AMAX (OFF_NATT + N_NODES * 4)        // N*2 (uint ordered-keys)
#define OFF_ASUM (OFF_AMAX + N_NODES * 2)        // N*2
#define OFF_SUP  (OFF_ASUM + N_NODES * 2)        // N*64
#define ZERO_WORDS (N_NODES * 68)                // amax + asum + support contiguous

#define M_TILES 3125                             // 50000 / 16

// ---------------- helpers ----------------
__device__ __forceinline__ float wred32(float v) {
#pragma unroll
  for (int m = 16; m >= 1; m >>= 1) v += __shfl_xor(v, m, 32);
  return v;
}

__device__ __forceinline__ unsigned ordkey(float f) {
  unsigned u = __float_as_uint(f);
  return (u & 0x80000000u) ? ~u : (u | 0x80000000u);
}
__device__ __forceinline__ float ordval(unsigned m) {
  return __uint_as_float((m & 0x80000000u) ? (m & 0x7FFFFFFFu) : ~m);
}
__device__ __forceinline__ float lrelu02(float a) { return a >= 0.0f ? a : 0.2f * a; }

// ---------------- kernels ----------------
__global__ void zero_u32(unsigned* __restrict__ p, int n) {
  const int idx = blockIdx.x * blockDim.x + threadIdx.x;
  if (idx < n) p[idx] = 0u;
}

// Pair-interleaved W^T: wt2[(k>>1)*256 + o*2 + (k&1)] = W[o][k]
// -> a lane's B fragment (K, K+1) for column o is one contiguous float2.
__global__ void transpose128(const float* __restrict__ w, float* __restrict__ wt2) {
  const int idx = blockIdx.x * blockDim.x + threadIdx.x;
  if (idx < 128 * 128) {
    const int o = idx >> 7, k = idx & 127;
    wt2[(k >> 1) * 256 + o * 2 + (k & 1)] = w[idx];
  }
}

// mx[N x 128] = X[N x 128] @ W^T  via fp32 WMMA (16x16x4).
// One wave -> one 16x128 block: 8 x v8f accumulators (64 VGPRs, no spill).
__global__ __launch_bounds__(256) void gemm_wmma_f32(const float* __restrict__ x,
                                                     const float* __restrict__ wt2,
                                                     float* __restrict__ mx) {
  const int wid  = (blockIdx.x * blockDim.x + threadIdx.x) >> 5;
  const int lane = threadIdx.x & 31;
  if (wid >= M_TILES) return;
  const int m0   = wid * 16;
  const int half = lane >> 4;               // 0: K+0/1, 1: K+2/3 (A & B layout halves)
  const int l15  = lane & 15;

  v8f acc[8];
#pragma unroll
  for (int j = 0; j < 8; ++j) acc[j] = (v8f){0.f,0.f,0.f,0.f,0.f,0.f,0.f,0.f};

  const float* ap = x + (size_t)(m0 + l15) * 128 + half * 2;

  for (int k0 = 0; k0 < 128; k0 += 4) {
    const v2f a = *(const v2f*)(ap + k0);
    const int pair = (k0 >> 1) + half;      // (k0 + 2*half) / 2
    const float* wp = wt2 + (size_t)pair * 256 + l15 * 2;
#pragma unroll
    for (int j = 0; j < 8; ++j) {
      const v2f b = *(const v2f*)(wp + j * 32);   // (W^T[kr][o], W^T[kr+1][o])
      acc[j] = __builtin_amdgcn_wmma_f32_16x16x4_f32(false, a, false, b,
                                                     (short)0, acc[j], false, false);
    }
  }

#pragma unroll
  for (int j = 0; j < 8; ++j)
#pragma unroll
    for (int v = 0; v < 8; ++v) {
      const int row = m0 + v + 8 * half;    // C layout: vgpr v -> M=v (+8 for hi half)
      mx[(size_t)row * 128 + j * 16 + l15] = acc[j][v];
    }
}

// One wave per node: mobius_matvec rescale -> proj -> mobius_add(hyp_bias) -> proj
// -> logmap0, written in-place over mx; also emits per-node attention dots.
__global__ void row_transform(const float* __restrict__ x, float* __restrict__ hx,
                              const float* __restrict__ bias,
                              const float* __restrict__ att_i, const float* __restrict__ att_j,
                              float4* __restrict__ natt) {
  const int wid  = (blockIdx.x * blockDim.x + threadIdx.x) >> 5;
  const int lane = threadIdx.x & 31;
  if (wid >= N_NODES) return;

  const float4 xv = ((const float4*)(x + (size_t)wid * 128))[lane];
  float* hr = hx + (size_t)wid * 128;
  const float4 mv = ((const float4*)hr)[lane];
  const float4 bv = ((const float4*)bias)[lane];

  float sx = wred32(xv.x*xv.x + xv.y*xv.y + xv.z*xv.z + xv.w*xv.w);
  float sm = wred32(mv.x*mv.x + mv.y*mv.y + mv.z*mv.z + mv.w*mv.w);
  float sb = wred32(bv.x*bv.x + bv.y*bv.y + bv.z*bv.z + bv.w*bv.w);

  // mobius_matvec(c=1): res = tanh(|mx|/|x| * atanh(|x|)) * mx/|mx|
  const float xn  = fmaxf(sqrtf(sx), EPSF);
  const float mxn = fmaxf(sqrtf(sm), EPSF);
  const float t1  = tanhf((mxn / xn) * atanhf(fminf(xn, CLIPF)));
  const float s1  = t1 / mxn;
  const float s2  = (t1 > MAXNF) ? (MAXNF / t1) : 1.0f;   // proj
  const float sh  = s1 * s2;                              // h = sh * mv
  const float h2  = sh * sh * sm;                         // |h|^2

  // hyp_bias = proj(expmap0(bias))  (bias is zeros in practice; general path anyway)
  const float bn  = fmaxf(sqrtf(sb), EPSF);
  const float tb  = tanhf(bn);
  const float eb  = tb / bn;
  const float pb  = (tb > MAXNF) ? (MAXNF / tb) : 1.0f;
  const float sbb = eb * pb;                              // hb = sbb * bv
  const float hb2 = sbb * sbb * sb;

  const float dmb = wred32(mv.x*bv.x + mv.y*bv.y + mv.z*bv.z + mv.w*bv.w);
  const float xy  = sh * sbb * dmb;                       // <h, hb>

  // mobius_add(h, hb): p = ((1+2xy+|hb|^2) h + (1-|h|^2) hb) / (1+2xy+|h|^2|hb|^2)
  const float A   = 1.0f + 2.0f * xy + hb2;
  const float B   = 1.0f - h2;
  const float den = fmaxf(1.0f + 2.0f * xy + h2 * hb2, EPSF);
  const float ca  = A * sh / den;
  const float cb  = B * sbb / den;
  float4 pv;
  pv.x = ca * mv.x + cb * bv.x;  pv.y = ca * mv.y + cb * bv.y;
  pv.z = ca * mv.z + cb * bv.z;  pv.w = ca * mv.w + cb * bv.w;

  const float sp2 = wred32(pv.x*pv.x + pv.y*pv.y + pv.z*pv.z + pv.w*pv.w);
  const float pn  = fmaxf(sqrtf(sp2), EPSF);
  const float sj  = (pn > MAXNF) ? (MAXNF / pn) : 1.0f;   // proj
  const float pn2 = pn * sj;
  const float sl  = atanhf(fminf(pn2, CLIPF)) / fmaxf(pn2, EPSF); // logmap0
  const float st  = sj * sl;

  float4 tv;
  tv.x = st * pv.x;  tv.y = st * pv.y;  tv.z = st * pv.z;  tv.w = st * pv.w;
  ((float4*)hr)[lane] = tv;                                // x_t0 row, in-place

  // per-node attention dots (head0 = lanes 0..15 / cols 0..63, head1 = lanes 16..31)
  const float4 aiv = ((const float4*)att_i)[lane];
  const float4 ajv = ((const float4*)att_j)[lane];
  float di = tv.x*aiv.x + tv.y*aiv.y + tv.z*aiv.z + tv.w*aiv.w;
  float dj = tv.x*ajv.x + tv.y*ajv.y + tv.z*ajv.z + tv.w*ajv.w;
#pragma unroll
  for (int m = 8; m >= 1; m >>= 1) { di += __shfl_xor(di, m, 32); dj += __shfl_xor(dj, m, 32); }
  const float di1 = __shfl(di, 16, 32);
  const float dj1 = __shfl(dj, 16, 32);
  if (lane == 0) natt[wid] = make_float4(di, di1, dj, dj1);
}

__global__ void edge_logit_max(const int* __restrict__ ei, const float4* __restrict__ natt,
                               unsigned* __restrict__ amax) {
  const int e = blockIdx.x * blockDim.x + threadIdx.x;
  if (e >= ET) return;
  int i, j;
  if (e < E_EDGES) { i = ei[e]; j = ei[E_EDGES + e]; } else { i = e - E_EDGES; j = i; }
  const float4 ai = natt[i];
  const float4 aj = natt[j];
  atomicMax(&amax[2 * i + 0], ordkey(lrelu02(ai.x + aj.z)));
  atomicMax(&amax[2 * i + 1], ordkey(lrelu02(ai.y + aj.w)));
}

__global__ void edge_exp_sum(const int* __restrict__ ei, const float4* __restrict__ natt,
                             const unsigned* __restrict__ amax, float* __restrict__ asum) {
  const int e = blockIdx.x * blockDim.x + threadIdx.x;
  if (e >= ET) return;
  int i, j;
  if (e < E_EDGES) { i = ei[e]; j = ei[E_EDGES + e]; } else { i = e - E_EDGES; j = i; }
  const float4 ai = natt[i];
  const float4 aj = natt[j];
  atomicAdd(&asum[2 * i + 0], expf(lrelu02(ai.x + aj.z) - ordval(amax[2 * i + 0])));
  atomicAdd(&asum[2 * i + 1], expf(lrelu02(ai.y + aj.w) - ordval(amax[2 * i + 1])));
}

// One wave per edge: support[i,c] += 0.5*(w0*xt0[j,c] + w1*xt0[j,64+c])  (mean folded)
__global__ void edge_scatter(const int* __restrict__ ei, const float4* __restrict__ natt,
                             const unsigned* __restrict__ amax, const float* __restrict__ asum,
                             const float* __restrict__ hx, float* __restrict__ support) {
  const int wid  = (blockIdx.x * blockDim.x + threadIdx.x) >> 5;
  const int lane = threadIdx.x & 31;
  if (wid >= ET) return;
  int i, j;
  if (wid < E_EDGES) { i = ei[wid]; j = ei[E_EDGES + wid]; } else { i = wid - E_EDGES; j = i; }
  const float4 ai = natt[i];
  const float4 aj = natt[j];
  const float w0 = 0.5f * expf(lrelu02(ai.x + aj.z) - ordval(amax[2 * i + 0]))
                        / fmaxf(asum[2 * i + 0], EPSF);
  const float w1 = 0.5f * expf(lrelu02(ai.y + aj.w) - ordval(amax[2 * i + 1]))
                        / fmaxf(asum[2 * i + 1], EPSF);
  const float2* xr = (const float2*)(hx + (size_t)j * 128);
  const float2 h0 = xr[lane];          // cols 2*lane      (head 0)
  const float2 h1 = xr[32 + lane];     // cols 64 + 2*lane (head 1)
  float* outp = support + (size_t)i * 64 + 2 * lane;
  atomicAdd(outp + 0, w0 * h0.x + w1 * h1.x);
  atomicAdd(outp + 1, w0 * h0.y + w1 * h1.y);
}

// One wave per node: out = proj(expmap0(leaky_relu(logmap0(proj(expmap0(mean))), 0.01)))
__global__ void finalize(const float* __restrict__ support, float* __restrict__ out) {
  const int wid  = (blockIdx.x * blockDim.x + threadIdx.x) >> 5;
  const int lane = threadIdx.x & 31;
  if (wid >= N_NODES) return;

  const float2 m = ((const float2*)(support + (size_t)wid * 64))[lane];
  const float s2m = wred32(m.x * m.x + m.y * m.y);
  const float mn  = fmaxf(sqrtf(s2m), EPSF);
  const float tn  = tanhf(mn);
  const float e1  = tn / mn;                               // expmap0
  const float p1  = (tn > MAXNF) ? (MAXNF / tn) : 1.0f;    // proj
  const float pn  = tn * p1;
  const float sl  = atanhf(fminf(pn, CLIPF)) / fmaxf(pn, EPSF);  // logmap0
  const float s   = e1 * p1 * sl;

  float tx = s * m.x, ty = s * m.y;
  tx = tx >= 0.0f ? tx : 0.01f * tx;                       // leaky_relu(0.01)
  ty = ty >= 0.0f ? ty : 0.01f * ty;

  const float s2t = wred32(tx * tx + ty * ty);
  const float xn2 = fmaxf(sqrtf(s2t), EPSF);
  const float tn2 = tanhf(xn2);
  const float e2  = tn2 / xn2;                             // expmap0
  const float p2  = (tn2 > MAXNF) ? (MAXNF / tn2) : 1.0f;  // proj
  const float st  = e2 * p2;

  float* op = out + (size_t)wid * 64 + 2 * lane;
  op[0] = st * tx;
  op[1] = st * ty;
}

// ---------------- host launcher ----------------
extern "C" void kernel_launch(void* const* d_in, const int* in_sizes, int n_in,
                              void* d_out, int out_size, void* d_ws, size_t ws_size,
                              hipStream_t stream) {
  const float* x     = (const float*)d_in[0];
  const int*   ei    = (const int*)d_in[1];
  const float* w     = (const float*)d_in[2];
  const float* bias  = (const float*)d_in[3];
  const float* att_i = (const float*)d_in[4];
  const float* att_j = (const float*)d_in[5];
  float* out = (float*)d_out;
  float* ws  = (float*)d_ws;

  float*    wt2     = ws + OFF_WT;
  float*    hx      = ws + OFF_HX;
  float4*   natt    = (float4*)(ws + OFF_NATT);
  unsigned* amax    = (unsigned*)(ws + OFF_AMAX);
  float*    asum    = ws + OFF_ASUM;
  float*    support = ws + OFF_SUP;

  // 0) zero amax/asum/support (contiguous region)
  zero_u32<<<(ZERO_WORDS + 255) / 256, 256, 0, stream>>>((unsigned*)(ws + OFF_AMAX), ZERO_WORDS);
  // 1) pair-interleaved W^T for single-b64 WMMA B-fragment loads
  transpose128<<<64, 256, 0, stream>>>(w, wt2);
  // 2) fp32 WMMA GEMM: hx = X @ W^T   (3125 16-row tiles, 8 waves/block)
  gemm_wmma_f32<<<(M_TILES + 7) / 8, 256, 0, stream>>>(x, wt2, hx);
  // 3) hyperbolic row transform (in-place) + per-node attention dots
  row_transform<<<(N_NODES + 7) / 8, 256, 0, stream>>>(x, hx, bias, att_i, att_j, natt);
  // 4) edge softmax: segment max, then exp-sum
  edge_logit_max<<<(ET + 255) / 256, 256, 0, stream>>>(ei, natt, amax);
  edge_exp_sum<<<(ET + 255) / 256, 256, 0, stream>>>(ei, natt, amax, asum);
  // 5) weighted scatter (mean over heads folded in)
  edge_scatter<<<(ET + 7) / 8, 256, 0, stream>>>(ei, natt, amax, asum, hx, support);
  // 6) final hyperbolic activation chain
  finalize<<<(N_NODES + 7) / 8, 256, 0, stream>>>(support, out);
}